// Graph_Learn_240518168582
// MI455X (gfx1250) — compile-verified
//
#include <hip/hip_runtime.h>
#include <hip/hip_bf16.h>

typedef float v2f __attribute__((ext_vector_type(2)));
typedef float v8f __attribute__((ext_vector_type(8)));

#define ALPHA 0.0001f
#define NB 16      // batches
#define TDIM 8     // time dim of x; we use slice T/2 = 4
#define V 256      // vertices
#define F 64       // features
#define FP 65      // padded LDS row stride -> bank (j+f)&63, conflict-free

#define S_ELEMS (NB * V * V)   // 1,048,576 floats; then Sloss, dloss

// ---------------------------------------------------------------------------
// Zero the two scalar accumulators at the tail of d_out.
// ---------------------------------------------------------------------------
__global__ void init_scalars_kernel(float* __restrict__ out) {
    if (threadIdx.x == 0) {
        out[S_ELEMS]     = 0.0f;  // Sloss
        out[S_ELEMS + 1] = 0.0f;  // dloss
    }
}

// ---------------------------------------------------------------------------
// Kernel 1: per-batch fused scores / softmax-over-i / Sloss + s1,s2 sums.
// grid = 16 blocks (one per n), block = 256 threads (8 wave32).
// Thread j owns column j of S[n] so the axis=1 normalization is thread-local.
// ---------------------------------------------------------------------------
__global__ __launch_bounds__(256) void graph_scores_kernel(
    const float* __restrict__ x, const float* __restrict__ a,
    float* __restrict__ out, float* __restrict__ ws)
{
    __shared__ float lx[V * FP];   // xm[n] staged, padded rows
    __shared__ float la[F];
    __shared__ float red[256];

    const int n   = blockIdx.x;
    const int tid = threadIdx.x;
    const float* __restrict__ xm = x + (size_t)(n * TDIM + TDIM / 2) * V * F;

    if (tid < F) la[tid] = a[tid];
    // coalesced global load of 256x64, store padded into LDS
    for (int idx = tid; idx < V * F; idx += 256) {
        const int r = idx >> 6, f = idx & 63;
        lx[r * FP + f] = xm[idx];
    }
    __syncthreads();

    // s1[f] = sum_j xm[j,f], s2[f] = sum_j xm[j,f]^2  -> workspace for kernel 2
    if (tid < F) {
        float s1 = 0.0f, s2 = 0.0f;
        for (int j = 0; j < V; ++j) {
            const float v = lx[j * FP + tid];   // lanes hit distinct banks
            s1 += v; s2 += v * v;
        }
        ws[n * 2 * F + tid]     = s1;
        ws[n * 2 * F + F + tid] = s2;
    }
    __syncthreads();

    // Pass 1: tmpS column j, accumulate column sum (normalizer over i).
    const int j = tid;
    float* __restrict__ Sn = out + (size_t)n * V * V;
    float colsum = 0.0f;
    for (int i = 0; i < V; ++i) {
        const float* __restrict__ xi = &lx[i * FP];  // broadcast across lanes
        const float* __restrict__ xj = &lx[j * FP];  // conflict-free (padded)
        float acc = 0.0f;
#pragma unroll 8
        for (int f = 0; f < F; ++f)
            acc = fmaf(la[f], fabsf(xi[f] - xj[f]), acc);
        const float t = __expf(fmaxf(acc, 0.0f));    // relu then exp; t >= 1
        Sn[(size_t)i * V + j] = t;                   // coalesced across j
        colsum += t;
    }

    // Pass 2: normalize in place (coalesced RMW) + accumulate sum(S^2).
    const float inv = 1.0f / colsum;                 // colsum >= 256, safe
    float sq = 0.0f;
    for (int i = 0; i < V; ++i) {
        const float s = Sn[(size_t)i * V + j] * inv;
        Sn[(size_t)i * V + j] = s;
        sq += s * s;
    }

    red[tid] = sq;
    __syncthreads();
    for (int off = 128; off > 0; off >>= 1) {
        if (tid < off) red[tid] += red[tid + off];
        __syncthreads();
    }
    if (tid == 0)
        atomicAdd(&out[S_ELEMS], (ALPHA / (float)NB) * red[0]);
}

// ---------------------------------------------------------------------------
// Kernel 2: dloss = ALPHA * sum(S @ d2) via V_WMMA_F32_16X16X4_F32.
// One wave32 per 16x16 output tile; K = 256 in steps of 4 (64 WMMA/wave).
// d2[n,k,f] = 256*x^2 - 2*x*s1[f] + s2[f] reconstructed on the fly (no 1MB
// d2 buffer). A layout (16x4 f32): lanes 0-15 -> K=0,1 ; lanes 16-31 -> K=2,3.
// B (4x16) mirrors it; C/D: 8 VGPRs, lanes 0-15 M=r, lanes 16-31 M=r+8.
// ---------------------------------------------------------------------------
__global__ __launch_bounds__(32) void dloss_wmma_kernel(
    const float* __restrict__ x, const float* __restrict__ S,
    const float* __restrict__ ws, float* __restrict__ dloss)
{
    const int lane = threadIdx.x;
    const int m    = lane & 15;   // row of A / col(f) of B this lane serves
    const int kh   = lane >> 4;   // K-half selector
    const int f0   = blockIdx.x * 16;
    const int i0   = blockIdx.y * 16;
    const int n    = blockIdx.z;

    const float* __restrict__ Srow =
        S + (size_t)n * V * V + (size_t)(i0 + m) * V;
    const float* __restrict__ xmcol =
        x + (size_t)(n * TDIM + TDIM / 2) * V * F + (f0 + m);
    const float s1f = ws[n * 2 * F + (f0 + m)];
    const float s2f = ws[n * 2 * F + F + (f0 + m)];

    v8f c = {};
    for (int k0 = 0; k0 < V; k0 += 4) {
        const int k = k0 + 2 * kh;
        v2f av, bv;
        av[0] = Srow[k];
        av[1] = Srow[k + 1];
        const float v0 = xmcol[(size_t)k * F];
        const float v1 = xmcol[(size_t)(k + 1) * F];
        bv[0] = fmaf(256.0f * v0 - 2.0f * s1f, v0, s2f);
        bv[1] = fmaf(256.0f * v1 - 2.0f * s1f, v1, s2f);
        c = __builtin_amdgcn_wmma_f32_16x16x4_f32(
                /*neg_a=*/false, av, /*neg_b=*/false, bv,
                /*c_mod=*/(short)0, c, /*reuse_a=*/false, /*reuse_b=*/false);
    }

    // Tile-wide sum: 8 accum VGPRs per lane cover all 256 tile elements.
    float sum = c[0] + c[1] + c[2] + c[3] + c[4] + c[5] + c[6] + c[7];
    for (int off = 16; off > 0; off >>= 1)
        sum += __shfl_xor(sum, off, 32);
    if (lane == 0)
        atomicAdd(dloss, ALPHA * sum);
}

// ---------------------------------------------------------------------------
extern "C" void kernel_launch(void* const* d_in, const int* in_sizes, int n_in,
                              void* d_out, int out_size, void* d_ws, size_t ws_size,
                              hipStream_t stream) {
    const float* x = (const float*)d_in[0];   // (16, 8, 256, 64) f32
    const float* a = (const float*)d_in[1];   // (64, 1) f32
    float* out = (float*)d_out;               // [S | Sloss | dloss]
    float* ws  = (float*)d_ws;                // s1,s2 per (n,f): 8 KB

    init_scalars_kernel<<<1, 32, 0, stream>>>(out);
    graph_scores_kernel<<<NB, 256, 0, stream>>>(x, a, out, ws);
    dloss_wmma_kernel<<<dim3(F / 16, V / 16, NB), 32, 0, stream>>>(
        x, out, ws, out + S_ELEMS + 1);
}